// Agent_86620900426118
// MI455X (gfx1250) — compile-verified
//
#include <hip/hip_runtime.h>

typedef __attribute__((ext_vector_type(2))) float v2f;
typedef __attribute__((ext_vector_type(8))) float v8f;

#define NP    16
#define NF    512
#define NR    36
#define MUL0  128
#define MUL1  64
#define DIM   320
#define KDIM  192          // 3*MUL1
#define NPR   (NP*NR)      // 576
#define NEGINF (-1000000000.0f)
#define ALPHA  0.009021097956f   // 1/sqrt(3*64*64) = sqrt(3)/192
#define INV_SQRT_MUL0 0.08838834765f // 1/sqrt(128)

// ---------------------------------------------------------------------------
// Prep A: X1[p*36+r][u*3+m] = sum_n R3[r][m][n] * feat[p][128 + u*3 + n]
// R3[r][m][n] is read from the (u=v=0) block of d_mats: d_mats[r, 128+m, 128+n]
// ---------------------------------------------------------------------------
__global__ __launch_bounds__(256) void Agent_prep_x1(
    const float* __restrict__ features, const float* __restrict__ d_mats,
    float* __restrict__ X1) {
  int t = blockIdx.x * 256 + threadIdx.x;
  if (t >= NPR * KDIM) return;
  int pr = t / KDIM, c = t - pr * KDIM;
  int u = c / 3, mm = c - u * 3;
  int p = pr / NR, r = pr - p * NR;
  const float* R3row = d_mats + (size_t)r * DIM * DIM + (size_t)(MUL0 + mm) * DIM + MUL0;
  const float* fe = features + p * DIM + MUL0 + u * 3;
  X1[t] = R3row[0] * fe[0] + R3row[1] * fe[1] + R3row[2] * fe[2];
}

// ---------------------------------------------------------------------------
// Prep B: Y2[f][u*3+m] = alpha * sum_v W_tp[u][v] * frag[f][128 + v*3 + m]
// ---------------------------------------------------------------------------
__global__ __launch_bounds__(256) void Agent_prep_y2(
    const float* __restrict__ frag, const float* __restrict__ w_tp,
    float* __restrict__ Y2) {
  int t = blockIdx.x * 256 + threadIdx.x;
  if (t >= NF * KDIM) return;
  int f = t / KDIM, c = t - f * KDIM;
  int u = c / 3, mm = c - u * 3;
  const float* fr = frag + f * DIM + MUL0 + mm;  // stride 3 over v
  const float* w  = w_tp + u * MUL1;
  float s = 0.0f;
  #pragma unroll 8
  for (int v = 0; v < MUL1; ++v) s += w[v] * fr[3 * v];
  Y2[t] = s * ALPHA;
}

// ---------------------------------------------------------------------------
// GEMM: L[pr][f] = X1[pr][:] . Y2[f][:]   (M=576, N=512, K=192), fp32 WMMA.
// One wave per 16x16 tile; 48 x V_WMMA_F32_16X16X4_F32 per tile.
// A lane L: m = L&15, k-pair = 2*(L>>4)        -> float2 from X1 row
// B lane L: n = L&15, k-pair = 2*(L>>4)        -> float2 from Y2 row (B = Y2^T)
// D vgpr i: row = 16*tm + 8*(L>>4) + i, col = 16*tn + (L&15)
// ---------------------------------------------------------------------------
__global__ __launch_bounds__(256) void Agent_gemm_wmma(
    const float* __restrict__ X1, const float* __restrict__ Y2,
    float* __restrict__ L) {
  const int wave = (blockIdx.x << 3) + (threadIdx.x >> 5);
  const int lane = threadIdx.x & 31;
  const int tm = wave >> 5;        // 0..35  (N/16 = 32 tiles)
  const int tn = wave & 31;        // 0..31
  const int l15 = lane & 15;
  const int kh = lane >> 4;        // 0/1 -> k-pair select

  const float* arow = X1 + (size_t)(tm * 16 + l15) * KDIM + 2 * kh;
  const float* brow = Y2 + (size_t)(tn * 16 + l15) * KDIM + 2 * kh;

  v8f c = {0.f, 0.f, 0.f, 0.f, 0.f, 0.f, 0.f, 0.f};
  #pragma unroll 12
  for (int k = 0; k < KDIM; k += 4) {
    v2f a = *(const v2f*)(arow + k);
    v2f b = *(const v2f*)(brow + k);
    c = __builtin_amdgcn_wmma_f32_16x16x4_f32(false, a, false, b,
                                              (short)0, c, false, false);
  }

  float* drow = L + (size_t)(tm * 16 + kh * 8) * NF + tn * 16 + l15;
  #pragma unroll
  for (int i = 0; i < 8; ++i) drow[(size_t)i * NF] = c[i];
}

// ---------------------------------------------------------------------------
// Finalize: per-p masked log-softmax over F*R, gather logprob, entropy, value.
// logits[p, f*R + r] = L[p*R + r][f]
// ---------------------------------------------------------------------------
__global__ __launch_bounds__(256) void Agent_finalize(
    const float* __restrict__ L, const float* __restrict__ features,
    const unsigned char* __restrict__ masks, const int* __restrict__ actions,
    const float* __restrict__ w_critic, float* __restrict__ out) {
  __shared__ float sh[256];
  __shared__ float sh2[256];
  const int p = blockIdx.x;
  const int tid = threadIdx.x;
  const int NE = NF * NR;  // 18432

  // pass 1: max of masked logits
  float m = NEGINF;
  for (int e = tid; e < NE; e += 256) {
    int f = e / NR, r = e - f * NR;
    float x = masks[p * NF + f] ? L[(size_t)(p * NR + r) * NF + f] : NEGINF;
    m = fmaxf(m, x);
  }
  sh[tid] = m;
  __syncthreads();
  for (int s = 128; s > 0; s >>= 1) {
    if (tid < s) sh[tid] = fmaxf(sh[tid], sh[tid + s]);
    __syncthreads();
  }
  const float M = sh[0];
  __syncthreads();

  // pass 2: Z = sum exp(x-M);  S = sum_masked (x-M)*exp(x-M)
  float z = 0.0f, ssum = 0.0f;
  for (int e = tid; e < NE; e += 256) {
    int f = e / NR, r = e - f * NR;
    bool mk = masks[p * NF + f] != 0;
    float x = mk ? L[(size_t)(p * NR + r) * NF + f] : NEGINF;
    float t = x - M;
    float et = expf(t);
    z += et;
    if (mk) ssum += t * et;
  }
  sh[tid] = z;
  sh2[tid] = ssum;
  __syncthreads();
  for (int s = 128; s > 0; s >>= 1) {
    if (tid < s) { sh[tid] += sh[tid + s]; sh2[tid] += sh2[tid + s]; }
    __syncthreads();
  }
  const float Z = sh[0], S = sh2[0];
  __syncthreads();

  // value: feat[p,:128] . w_critic / sqrt(128)
  sh[tid] = (tid < MUL0) ? features[p * DIM + tid] * w_critic[tid] : 0.0f;
  __syncthreads();
  for (int s = 128; s > 0; s >>= 1) {
    if (tid < s) sh[tid] += sh[tid + s];
    __syncthreads();
  }

  if (tid == 0) {
    float logZ = logf(Z);
    int a = actions[p];
    int f = a / NR, r = a - f * NR;
    float xa = masks[p * NF + f] ? L[(size_t)(p * NR + r) * NF + f] : NEGINF;
    out[p]          = xa - M - logZ;      // frag_logprob
    out[NP + p]     = logZ - S / Z;       // entropy
    out[2 * NP + p] = sh[0] * INV_SQRT_MUL0;  // value
  }
}

// ---------------------------------------------------------------------------
extern "C" void kernel_launch(void* const* d_in, const int* in_sizes, int n_in,
                              void* d_out, int out_size, void* d_ws, size_t ws_size,
                              hipStream_t stream) {
  const float*         features = (const float*)d_in[0];
  const float*         frag     = (const float*)d_in[1];
  const unsigned char* masks    = (const unsigned char*)d_in[2];
  const int*           actions  = (const int*)d_in[3];
  const float*         d_mats   = (const float*)d_in[4];
  const float*         w_tp     = (const float*)d_in[5];
  const float*         w_critic = (const float*)d_in[6];
  float* ws = (float*)d_ws;

  float* X1 = ws;                      // 576*192  = 110592 floats
  float* Y2 = X1 + NPR * KDIM;         // 512*192  =  98304 floats
  float* L  = Y2 + NF * KDIM;          // 576*512  = 294912 floats (~2 MB total)

  Agent_prep_x1<<<(NPR * KDIM + 255) / 256, 256, 0, stream>>>(features, d_mats, X1);
  Agent_prep_y2<<<(NF * KDIM + 255) / 256, 256, 0, stream>>>(frag, w_tp, Y2);
  // (576/16)*(512/16) = 1152 waves = 144 blocks of 8 waves
  Agent_gemm_wmma<<<144, 256, 0, stream>>>(X1, Y2, L);
  Agent_finalize<<<NP, 256, 0, stream>>>(L, features, masks, actions, w_critic,
                                         (float*)d_out);
}